// PairRE_19370302505584
// MI455X (gfx1250) — compile-verified
//
#include <hip/hip_runtime.h>
#include <math.h>

typedef float v2f __attribute__((ext_vector_type(2)));
typedef float v8f __attribute__((ext_vector_type(8)));

#define EDIM   256
#define KCAT   512          // concatenated K = 2*EDIM
#define BATCH  512
#define NNEG   2048

// ---------------------------------------------------------------------------
// Prep pass 1: one block per batch row.
//   P[b][0:256]    = rt^2
//   P[b][256:512]  = -2 * rt * hh        (hh = normalize(head)*rh)
//   Avec[b]        = sum(hh^2)
// ---------------------------------------------------------------------------
__global__ void __launch_bounds__(EDIM)
prep_head(const float* __restrict__ head, const float* __restrict__ rel,
          const int* __restrict__ rid, float* __restrict__ P,
          float* __restrict__ Avec) {
  __shared__ float red[EDIM];
  const int b = blockIdx.x;
  const int e = threadIdx.x;

  const float h = head[b * EDIM + e];
  red[e] = h * h;
  __syncthreads();
  for (int s = EDIM / 2; s > 0; s >>= 1) {
    if (e < s) red[e] += red[e + s];
    __syncthreads();
  }
  const float inv = 1.0f / fmaxf(sqrtf(red[0]), 1e-12f);
  __syncthreads();

  const float* r = rel + (size_t)rid[b] * (2 * EDIM);
  const float rh = r[e];
  const float rt = r[EDIM + e];
  const float hh = h * inv * rh;

  P[(size_t)b * KCAT + e]        = rt * rt;
  P[(size_t)b * KCAT + EDIM + e] = -2.0f * rt * hh;

  red[e] = hh * hh;
  __syncthreads();
  for (int s = EDIM / 2; s > 0; s >>= 1) {
    if (e < s) red[e] += red[e + s];
    __syncthreads();
  }
  if (e == 0) Avec[b] = red[0];
}

// ---------------------------------------------------------------------------
// Prep pass 2: one block per negative-tail row (B matrix, stored transposed:
// K contiguous so the WMMA B-fragment is one aligned float2 load).
//   T[n][0:256]   = tn^2
//   T[n][256:512] = tn
// ---------------------------------------------------------------------------
__global__ void __launch_bounds__(EDIM)
prep_tail(const float* __restrict__ tail, float* __restrict__ T) {
  __shared__ float red[EDIM];
  const int n = blockIdx.x;
  const int e = threadIdx.x;

  const float t = tail[(size_t)n * EDIM + e];
  red[e] = t * t;
  __syncthreads();
  for (int s = EDIM / 2; s > 0; s >>= 1) {
    if (e < s) red[e] += red[e + s];
    __syncthreads();
  }
  const float inv = 1.0f / fmaxf(sqrtf(red[0]), 1e-12f);

  const float tn = t * inv;
  T[(size_t)n * KCAT + e]        = tn * tn;
  T[(size_t)n * KCAT + EDIM + e] = tn;
}

// ---------------------------------------------------------------------------
// GEMM + epilogue: out[b][n] = -sqrt(max(P·T^T + Avec[b], 0))
// 8 waves / block; each wave owns a 32(M) x 64(N) tile = 2x4 f32 WMMA accs.
// V_WMMA_F32_16X16X4_F32 fragment layout (ISA 7.12.2):
//   A/B: lane<16 -> row/col = lane, holds {K, K+1}; lane>=16 -> {K+2, K+3}
//   C/D: N = lane%16, M = vgpr + 8*(lane/16)
// ---------------------------------------------------------------------------
__global__ void __launch_bounds__(256)
pairre_gemm(const float* __restrict__ P, const float* __restrict__ T,
            const float* __restrict__ Avec, float* __restrict__ out) {
  const int wave = (blockIdx.x * 256 + threadIdx.x) >> 5;  // 0..511
  const int lane = threadIdx.x & 31;
  const int half = lane >> 4;      // 0: K,K+1   1: K+2,K+3
  const int l16  = lane & 15;

  const int mt = wave & 15;        // 16 tiles of 32 rows  -> 512
  const int nt = wave >> 4;        // 32 tiles of 64 cols  -> 2048
  const int m0 = mt * 32;
  const int n0 = nt * 64;

  const float* pA0 = P + (size_t)(m0 + l16) * KCAT + 2 * half;
  const float* pA1 = P + (size_t)(m0 + 16 + l16) * KCAT + 2 * half;
  const float* pB0 = T + (size_t)(n0 + 0 * 16 + l16) * KCAT + 2 * half;
  const float* pB1 = T + (size_t)(n0 + 1 * 16 + l16) * KCAT + 2 * half;
  const float* pB2 = T + (size_t)(n0 + 2 * 16 + l16) * KCAT + 2 * half;
  const float* pB3 = T + (size_t)(n0 + 3 * 16 + l16) * KCAT + 2 * half;

  v8f acc[2][4] = {};

#pragma unroll 4
  for (int k = 0; k < KCAT; k += 4) {
    const v2f a0 = *(const v2f*)(pA0 + k);
    const v2f a1 = *(const v2f*)(pA1 + k);
    const v2f b0 = *(const v2f*)(pB0 + k);
    const v2f b1 = *(const v2f*)(pB1 + k);
    const v2f b2 = *(const v2f*)(pB2 + k);
    const v2f b3 = *(const v2f*)(pB3 + k);

    acc[0][0] = __builtin_amdgcn_wmma_f32_16x16x4_f32(false, a0, false, b0,
                                                      (short)0, acc[0][0], false, false);
    acc[0][1] = __builtin_amdgcn_wmma_f32_16x16x4_f32(false, a0, false, b1,
                                                      (short)0, acc[0][1], false, false);
    acc[0][2] = __builtin_amdgcn_wmma_f32_16x16x4_f32(false, a0, false, b2,
                                                      (short)0, acc[0][2], false, false);
    acc[0][3] = __builtin_amdgcn_wmma_f32_16x16x4_f32(false, a0, false, b3,
                                                      (short)0, acc[0][3], false, false);
    acc[1][0] = __builtin_amdgcn_wmma_f32_16x16x4_f32(false, a1, false, b0,
                                                      (short)0, acc[1][0], false, false);
    acc[1][1] = __builtin_amdgcn_wmma_f32_16x16x4_f32(false, a1, false, b1,
                                                      (short)0, acc[1][1], false, false);
    acc[1][2] = __builtin_amdgcn_wmma_f32_16x16x4_f32(false, a1, false, b2,
                                                      (short)0, acc[1][2], false, false);
    acc[1][3] = __builtin_amdgcn_wmma_f32_16x16x4_f32(false, a1, false, b3,
                                                      (short)0, acc[1][3], false, false);
  }

#pragma unroll
  for (int mi = 0; mi < 2; ++mi) {
#pragma unroll
    for (int ni = 0; ni < 4; ++ni) {
      const int col = n0 + ni * 16 + l16;
#pragma unroll
      for (int vg = 0; vg < 8; ++vg) {
        const int row = m0 + mi * 16 + half * 8 + vg;
        const float s = acc[mi][ni][vg] + Avec[row];
        out[(size_t)row * NNEG + col] = -sqrtf(fmaxf(s, 0.0f));
      }
    }
  }
}

// ---------------------------------------------------------------------------
extern "C" void kernel_launch(void* const* d_in, const int* in_sizes, int n_in,
                              void* d_out, int out_size, void* d_ws, size_t ws_size,
                              hipStream_t stream) {
  const float* head = (const float*)d_in[0];   // (512, 256)
  const float* tail = (const float*)d_in[1];   // (1, 2048, 256)
  const float* rel  = (const float*)d_in[2];   // (1000, 512)
  const int*   rid  = (const int*)d_in[3];     // (512,)
  float* out = (float*)d_out;                  // (512, 2048)

  float* P    = (float*)d_ws;                  // 512*512 f32   = 1 MB
  float* T    = P + (size_t)BATCH * KCAT;      // 2048*512 f32  = 4 MB
  float* Avec = T + (size_t)NNEG * KCAT;       // 512 f32

  prep_head<<<BATCH, EDIM, 0, stream>>>(head, rel, rid, P, Avec);
  prep_tail<<<NNEG, EDIM, 0, stream>>>(tail, T);

  // 512 waves total: 16 (M) x 32 (N) tiles of 32x64 -> 64 blocks of 8 waves
  pairre_gemm<<<64, 256, 0, stream>>>(P, T, Avec, out);
  (void)in_sizes; (void)n_in; (void)out_size; (void)ws_size;
}